// DualLearn2Proj_50027779064142
// MI455X (gfx1250) — compile-verified
//
#include <hip/hip_runtime.h>
#include <math.h>

typedef __attribute__((ext_vector_type(2))) float v2f;
typedef __attribute__((ext_vector_type(8))) float v8f;

#define BSZ   4096
#define OUTD  256
#define MDIM  128
#define KPAD  208   // 200 padded up to a multiple of 16

// ---------------------------------------------------------------------------
// CDNA5 async global->LDS copy (ASYNCcnt-tracked, no VGPR round trip).
// lds_byte = raw LDS byte offset (kernels use ONLY dynamic LDS => base 0).
// ---------------------------------------------------------------------------
__device__ __forceinline__ void async_ld16(unsigned lds_byte, const float* g) {
  asm volatile("global_load_async_to_lds_b128 %0, %1, off"
               :: "v"(lds_byte), "v"(g) : "memory");
}
__device__ __forceinline__ void wait_async_le2() {
  asm volatile("s_wait_asynccnt 0x2" ::: "memory");
}
__device__ __forceinline__ void wait_async_0() {
#if __has_builtin(__builtin_amdgcn_s_wait_asynccnt)
  __builtin_amdgcn_s_wait_asynccnt(0);
#else
  asm volatile("s_wait_asynccnt 0x0" ::: "memory");
#endif
}

// Zero-pad rows: dst[rows][Kdst] = src[rows][Ksrc] | 0
__global__ __launch_bounds__(256)
void k_pad(const float* __restrict__ src, float* __restrict__ dst,
           int rows, int Ksrc, int Kdst) {
  int idx = blockIdx.x * 256 + threadIdx.x;
  if (idx >= rows * Kdst) return;
  int r = idx / Kdst, c = idx - r * Kdst;
  dst[idx] = (c < Ksrc) ? src[(size_t)r * Ksrc + c] : 0.f;
}

// ---------------------------------------------------------------------------
// fp32 WMMA GEMM:  C[M,N] = act(A[M,K] @ W[N,K]^T + bias[N]),  K % 16 == 0.
// 256 thr (8 waves), 64x64 block tile, BK=16, double-buffered async staging.
// Dynamic LDS: 2 buffers x (As 64x20 + Ws 64x20) = 5120 floats.
// ---------------------------------------------------------------------------
template<int ACT>
__global__ __launch_bounds__(256)
void k_gemm(const float* __restrict__ A, const float* __restrict__ W,
            const float* __restrict__ bias, float* __restrict__ C,
            int N, int K) {
  extern __shared__ float smem[];
  const int t    = threadIdx.x;
  const int lane = t & 31, wv = t >> 5;
  const int half = lane >> 4, lm = lane & 15;
  const int wm = wv & 3, wn = wv >> 2;
  const int bm = blockIdx.y * 64, bn = blockIdx.x * 64;
  const int lrow = t >> 2;          // 0..63
  const int lseg = (t & 3) * 4;     // 0,4,8,12

  const float* gA = A + (size_t)(bm + lrow) * K + lseg;
  const float* gW = W + (size_t)(bn + lrow) * K + lseg;
  const unsigned sAs  = (unsigned)((lrow * 20 + lseg) * 4);  // bytes, buf0 As
  const unsigned sWs  = sAs + 1280u * 4u;
  const unsigned bufB = 2560u * 4u;                          // bytes per buffer

  const int S = K >> 4;
  async_ld16(sAs, gA);              // prologue: stage 0 -> buf0
  async_ld16(sWs, gW);

  v8f c0 = {}, c1 = {};
  for (int s = 0; s < S; ++s) {
    const int cur = s & 1;
    if (s + 1 < S) {                // prefetch next stage into other buffer
      const unsigned nb = (unsigned)(cur ^ 1) * bufB;
      async_ld16(sAs + nb, gA + (size_t)(s + 1) * 16);
      async_ld16(sWs + nb, gW + (size_t)(s + 1) * 16);
      wait_async_le2();             // in-order: stage s has landed
    } else {
      wait_async_0();
    }
    __syncthreads();
    const float* As = smem + cur * 2560;
    const float* Ws = As + 1280;
#pragma unroll
    for (int kk = 0; kk < 4; ++kk) {
      int col = kk * 4 + half * 2;
      v2f a  = *(const v2f*)&As[(wm * 16 + lm) * 20 + col];
      v2f b0 = *(const v2f*)&Ws[(wn * 32 + lm) * 20 + col];
      v2f b1 = *(const v2f*)&Ws[(wn * 32 + 16 + lm) * 20 + col];
      c0 = __builtin_amdgcn_wmma_f32_16x16x4_f32(false, a, false, b0, (short)0, c0, false, false);
      c1 = __builtin_amdgcn_wmma_f32_16x16x4_f32(false, a, false, b1, (short)0, c1, false, false);
    }
    __syncthreads();                // buffer reads done before next overwrite
  }
  const int n0 = bn + wn * 32 + lm;
  const int n1 = n0 + 16;
  const float bb0 = bias[n0], bb1 = bias[n1];
#pragma unroll
  for (int v = 0; v < 8; ++v) {
    int m = bm + wm * 16 + v + half * 8;
    float x0 = c0[v] + bb0;
    float x1 = c1[v] + bb1;
    if (ACT) { x0 = fmaxf(x0, 0.f); x1 = fmaxf(x1, 0.f); }
    C[(size_t)m * N + n0] = x0;
    C[(size_t)m * N + n1] = x1;
  }
}

// ---------------------------------------------------------------------------
// Row LayerNorm over 512 columns, in place. One block (256 thr) per row.
// ---------------------------------------------------------------------------
__global__ __launch_bounds__(256)
void k_ln(float* __restrict__ h, const float* __restrict__ g,
          const float* __restrict__ be) {
  __shared__ float rs[256], rss[256];
  const int t = threadIdx.x;
  float* row = h + (size_t)blockIdx.x * 512;
  v2f v = *(const v2f*)(row + t * 2);
  rs[t]  = v[0] + v[1];
  rss[t] = v[0] * v[0] + v[1] * v[1];
  __syncthreads();
  for (int off = 128; off > 0; off >>= 1) {
    if (t < off) { rs[t] += rs[t + off]; rss[t] += rss[t + off]; }
    __syncthreads();
  }
  float mu  = rs[0] * (1.f / 512.f);
  float var = rss[0] * (1.f / 512.f) - mu * mu;
  float r   = rsqrtf(var + 1e-5f);
  int c = t * 2;
  v2f o;
  o[0] = (v[0] - mu) * r * g[c] + be[c];
  o[1] = (v[1] - mu) * r * g[c + 1] + be[c + 1];
  *(v2f*)(row + c) = o;
}

// pz = concat(raw[:, :64], relu(raw[:, 64:])) in place; also copy into z0.
__global__ __launch_bounds__(256)
void k_pz_tail(float* __restrict__ pz, float* __restrict__ z0) {
  int idx = blockIdx.x * 256 + threadIdx.x;
  int col = idx & (OUTD - 1);
  float v = pz[idx];
  if (col >= 64) v = fmaxf(v, 0.f);
  pz[idx] = v;
  z0[idx] = v;
}

// Bias = b_eq @ WbProj^T; bscale = 1 + ||b_eq||; init flags/accumulator.
__global__ __launch_bounds__(256)
void k_bias_prep(const float* __restrict__ b_eq, const float* __restrict__ WbProj,
                 float* __restrict__ Bias, float* __restrict__ scal,
                 int* __restrict__ flags) {
  __shared__ float red[128];
  const int t = threadIdx.x;
  float s = 0.f;
  const float* wrow = WbProj + (size_t)t * MDIM;
  for (int m = 0; m < MDIM; ++m) s += b_eq[m] * wrow[m];
  Bias[t] = s;
  if (t < 128) red[t] = b_eq[t] * b_eq[t];
  __syncthreads();
  for (int off = 64; off > 0; off >>= 1) {
    if (t < off) red[t] += red[t + off];
    __syncthreads();
  }
  if (t == 0) {
    scal[0] = 1.f + sqrtf(red[0]);  // bscale
    scal[1] = 0.f;                  // crit accumulator
    flags[0] = 0;                   // converged flag
    flags[1] = 1;                   // curr_iter (starts at 1)
  }
}

// ---------------------------------------------------------------------------
// One fixed-point iteration. Block = 256 thr, owns 16 rows x all 256 cols.
// Phase 1: z_new = Bias + z @ Wz^T (ReLU tail), in place + kept in LDS.
// Phase 2: r = z_new @ A^T; accumulate sum of row 2-norms of (r - b_eq).
// Dynamic LDS: zs 16x260 | ws 256x20 | rowsq 16  = 9296 floats.
// ---------------------------------------------------------------------------
__global__ __launch_bounds__(256)
void k_iter_step(float* __restrict__ z, const float* __restrict__ Wz,
                 const float* __restrict__ Aop, const float* __restrict__ b_eq,
                 const float* __restrict__ Bias, float* __restrict__ accum,
                 const int* __restrict__ flags) {
  if (flags[0]) return;  // uniform early-out after convergence (graph-safe)
  extern __shared__ float smem[];
  float* zs    = smem;               // [16][260]
  float* ws    = smem + 16 * 260;    // [256][20]
  float* rowsq = ws + 256 * 20;      // [16]
  const unsigned wbase = (unsigned)(16 * 260 * 4);
  const int t    = threadIdx.x;
  const int lane = t & 31, wv = t >> 5;
  const int half = lane >> 4, lm = lane & 15;
  const int rb = blockIdx.x * 16;

  {  // async-load 16 full z rows into LDS
    int row = t >> 4;
    int col = (t & 15) * 16;
    const float* gz = z + (size_t)(rb + row) * OUTD + col;
    unsigned off = (unsigned)((row * 260 + col) * 4);
    async_ld16(off,      gz);
    async_ld16(off + 16, gz + 4);
    async_ld16(off + 32, gz + 8);
    async_ld16(off + 48, gz + 12);
  }

  // ---- phase 1: z @ Wz^T, K = 256, each wave covers 32 output cols ----
  v8f c0 = {}, c1 = {};
  for (int k0 = 0; k0 < OUTD; k0 += 16) {
    __syncthreads();                 // prior-stage ws reads done
    {  // stage Wz[n][k0..k0+15] for all 256 n
      const float* gw = Wz + (size_t)t * OUTD + k0;
      unsigned off = wbase + (unsigned)(t * 20 * 4);
      async_ld16(off,      gw);
      async_ld16(off + 16, gw + 4);
      async_ld16(off + 32, gw + 8);
      async_ld16(off + 48, gw + 12);
    }
    wait_async_0();                  // also covers the zs prologue asyncs
    __syncthreads();
#pragma unroll
    for (int kk = 0; kk < 4; ++kk) {
      int col = kk * 4 + half * 2;
      v2f a  = *(const v2f*)&zs[lm * 260 + k0 + col];
      v2f b0 = *(const v2f*)&ws[(wv * 32 + lm) * 20 + col];
      v2f b1 = *(const v2f*)&ws[(wv * 32 + 16 + lm) * 20 + col];
      c0 = __builtin_amdgcn_wmma_f32_16x16x4_f32(false, a, false, b0, (short)0, c0, false, false);
      c1 = __builtin_amdgcn_wmma_f32_16x16x4_f32(false, a, false, b1, (short)0, c1, false, false);
    }
  }
  __syncthreads();  // all frag reads of zs done before overwrite

  const int n0 = wv * 32 + lm;
  const int n1 = n0 + 16;
#pragma unroll
  for (int v = 0; v < 8; ++v) {
    int m = v + half * 8;
    float x0 = Bias[n0] + c0[v]; if (n0 >= 64) x0 = fmaxf(x0, 0.f);
    float x1 = Bias[n1] + c1[v]; if (n1 >= 64) x1 = fmaxf(x1, 0.f);
    zs[m * 260 + n0] = x0; zs[m * 260 + n1] = x1;
    z[(size_t)(rb + m) * OUTD + n0] = x0;
    z[(size_t)(rb + m) * OUTD + n1] = x1;
  }
  if (t < 16) rowsq[t] = 0.f;
  __syncthreads();

  // ---- phase 2: r = z_new @ A^T (128 cols), each wave one 16-col tile ----
  v8f c2 = {};
  for (int k0 = 0; k0 < OUTD; k0 += 16) {
    {  // stage A[n][k0..k0+15] for 128 n (2 threads per row)
      int row = t >> 1;
      int seg = (t & 1) * 8;
      const float* ga = Aop + (size_t)row * OUTD + k0 + seg;
      unsigned off = wbase + (unsigned)((row * 20 + seg) * 4);
      async_ld16(off,      ga);
      async_ld16(off + 16, ga + 4);
    }
    wait_async_0();
    __syncthreads();
#pragma unroll
    for (int kk = 0; kk < 4; ++kk) {
      int col = kk * 4 + half * 2;
      v2f a = *(const v2f*)&zs[lm * 260 + k0 + col];
      v2f b = *(const v2f*)&ws[(wv * 16 + lm) * 20 + col];
      c2 = __builtin_amdgcn_wmma_f32_16x16x4_f32(false, a, false, b, (short)0, c2, false, false);
    }
    __syncthreads();
  }

#pragma unroll
  for (int v = 0; v < 8; ++v) {
    int m = v + half * 8;
    int n = wv * 16 + lm;
    float d  = c2[v] - b_eq[n];
    float sq = d * d;                 // reduce over the 16 lanes sharing row m
    sq += __shfl_xor(sq, 1, 32);
    sq += __shfl_xor(sq, 2, 32);
    sq += __shfl_xor(sq, 4, 32);
    sq += __shfl_xor(sq, 8, 32);
    if (lm == 0) atomicAdd(&rowsq[m], sq);
  }
  __syncthreads();
  if (t < 16) rowsq[t] = sqrtf(rowsq[t]);
  __syncthreads();
  if (t == 0) {
    float s = 0.f;
    for (int i = 0; i < 16; ++i) s += rowsq[i];
    atomicAdd(accum, s);
  }
}

__global__ void k_iter_fin(float* scal, int* flags) {
  if (flags[0]) return;
  float crit = scal[1] / ((float)BSZ * scal[0]);
  flags[1] += 1;
  if (crit <= 1e-4f) flags[0] = 1;
  scal[1] = 0.f;
}

__global__ void k_write_iter(const int* flags, int* dst) { dst[0] = flags[1]; }

// ---------------------------------------------------------------------------
extern "C" void kernel_launch(void* const* d_in, const int* in_sizes, int n_in,
                              void* d_out, int out_size, void* d_ws, size_t ws_size,
                              hipStream_t stream) {
  (void)in_sizes; (void)n_in; (void)out_size; (void)ws_size;
  const float* x    = (const float*)d_in[0];
  const float* W0   = (const float*)d_in[1];
  const float* b0   = (const float*)d_in[2];
  const float* W1   = (const float*)d_in[3];
  const float* b1   = (const float*)d_in[4];
  const float* W2   = (const float*)d_in[5];
  const float* b2   = (const float*)d_in[6];
  const float* W3   = (const float*)d_in[7];
  const float* b3   = (const float*)d_in[8];
  const float* pW0  = (const float*)d_in[9];
  const float* pb0  = (const float*)d_in[10];
  const float* g0   = (const float*)d_in[11];
  const float* be0  = (const float*)d_in[12];
  const float* pW1  = (const float*)d_in[13];
  const float* pb1  = (const float*)d_in[14];
  const float* g1   = (const float*)d_in[15];
  const float* be1  = (const float*)d_in[16];
  const float* pWf  = (const float*)d_in[17];
  const float* pbf  = (const float*)d_in[18];
  const float* Aop  = (const float*)d_in[19];
  const float* b_eq = (const float*)d_in[20];
  const float* WzP  = (const float*)d_in[21];
  const float* WbP  = (const float*)d_in[22];

  float* out    = (float*)d_out;
  float* z_star = out;                            // [4096,256] (also z state)
  float* pz     = out + (size_t)BSZ * OUTD;       // [4096,256]
  int*   it_out = (int*)(out + (size_t)2 * BSZ * OUTD);

  float* f    = (float*)d_ws;
  float* xp   = f;                                // [4096,208] padded x
  float* w0p  = xp + (size_t)BSZ * KPAD;          // [512,208] padded W0
  float* act0 = w0p + (size_t)512 * KPAD;         // [4096,512]
  float* act1 = act0 + (size_t)BSZ * 512;         // [4096,512]
  float* Bias = act1 + (size_t)BSZ * 512;         // [256]
  float* scal = Bias + 256;                       // bscale, accum
  int*   flg  = (int*)(scal + 2);                 // converged, curr_iter

  dim3 blk(256);
  const size_t smG = 5120 * sizeof(float);        // k_gemm dynamic LDS
  const size_t smI = 9296 * sizeof(float);        // k_iter_step dynamic LDS

  // zero-pad K=200 -> 208 so async staging needs no tail path
  k_pad<<<(BSZ * KPAD + 255) / 256, blk, 0, stream>>>(x,  xp,  BSZ, 200, KPAD);
  k_pad<<<(512 * KPAD + 255) / 256, blk, 0, stream>>>(W0, w0p, 512, 200, KPAD);

  // optimality MLP
  k_gemm<1><<<dim3(8, 64), blk, smG, stream>>>(xp,   w0p, b0, act0, 512, KPAD);
  k_gemm<1><<<dim3(8, 64), blk, smG, stream>>>(act0, W1,  b1, act1, 512, 512);
  k_gemm<1><<<dim3(8, 64), blk, smG, stream>>>(act1, W2,  b2, act0, 512, 512);
  k_gemm<0><<<dim3(4, 64), blk, smG, stream>>>(act0, W3,  b3, act1, 256, 512);
  // projection net
  k_gemm<1><<<dim3(8, 64), blk, smG, stream>>>(act1, pW0, pb0, act0, 512, 256);
  k_ln<<<BSZ, blk, 0, stream>>>(act0, g0, be0);
  k_gemm<1><<<dim3(8, 64), blk, smG, stream>>>(act0, pW1, pb1, act1, 512, 512);
  k_ln<<<BSZ, blk, 0, stream>>>(act1, g1, be1);
  k_gemm<0><<<dim3(4, 64), blk, smG, stream>>>(act1, pWf, pbf, pz, 256, 512);
  k_pz_tail<<<BSZ, blk, 0, stream>>>(pz, z_star);
  // feasibility fixed-point loop (device-flag gated, graph-capture safe)
  k_bias_prep<<<1, blk, 0, stream>>>(b_eq, WbP, Bias, scal, flg);
  for (int i = 0; i < 20; ++i) {
    k_iter_step<<<BSZ / 16, blk, smI, stream>>>(z_star, WzP, Aop, b_eq, Bias, scal + 1, flg);
    k_iter_fin<<<1, 1, 0, stream>>>(scal, flg);
  }
  k_write_iter<<<1, 1, 0, stream>>>(flg, it_out);
}